// AdaptiveAugmentation_84061099918040
// MI455X (gfx1250) — compile-verified
//
#include <hip/hip_runtime.h>
#include <hip/hip_bf16.h>

typedef float v2f __attribute__((ext_vector_type(2)));
typedef float v8f __attribute__((ext_vector_type(8)));

#define HW   256
#define H1   511
#define W1   511
#define IMGS 192   // B*C = 64*3

// ---------------------------------------------------------------------------
// Stage 1: separable polyphase up-by-2 conv with sym6 low-pass (12 taps).
// y[o] = sum_t a[11-t] * x_dilated[o + t - 5], x_dilated[2i]=x[i].
// 32x32 output tile per block; 22x22 input patch in LDS; row pass then col pass.
// ---------------------------------------------------------------------------
__global__ __launch_bounds__(256) void up_lo_kernel(const float* __restrict__ x,
                                                    const float* __restrict__ lo2d,
                                                    float* __restrict__ y1) {
    __shared__ float sh_a[12];
    __shared__ float xt[22][24];
    __shared__ float ht[22][32];

    const int tid = threadIdx.x;
    if (tid < 12) sh_a[tid] = lo2d[tid] / sqrtf(lo2d[0]);  // 1D factor of outer(a,a)

    const int img  = blockIdx.x >> 8;        // 256 tiles per image
    const int rem  = blockIdx.x & 255;
    const int o_r0 = (rem >> 4) << 5;        // 16x16 grid of 32x32 tiles covers 511
    const int o_c0 = (rem & 15) << 5;
    const int IR0 = (o_r0 - 5) >> 1;         // floor((o0-5)/2): 22 input rows/cols
    const int IC0 = (o_c0 - 5) >> 1;

    const float* xs = x + (size_t)img * HW * HW;

    // Phase A: load (zero-padded) 22x22 input patch
    for (int idx = tid; idx < 22 * 22; idx += 256) {
        const int lr = idx / 22, lc = idx - lr * 22;
        const int gr = IR0 + lr, gc = IC0 + lc;
        float v = 0.0f;
        if (gr >= 0 && gr < HW && gc >= 0 && gc < HW) v = xs[gr * HW + gc];
        xt[lr][lc] = v;
    }
    __syncthreads();

    // Phase B: horizontal polyphase conv -> ht[22][32]
    for (int idx = tid; idx < 22 * 32; idx += 256) {
        const int ir = idx >> 5, oc = idx & 31;
        const int ocg = o_c0 + oc;
        const int t0  = (ocg + 1) & 1;                  // tap parity
        const int lib = ((ocg + t0 - 5) >> 1) - IC0;    // exact (even) division
        float acc = 0.0f;
#pragma unroll
        for (int tt = 0; tt < 6; ++tt)
            acc += sh_a[11 - t0 - 2 * tt] * xt[ir][lib + tt];
        ht[ir][oc] = acc;
    }
    __syncthreads();

    // Phase C: vertical polyphase conv -> y1 tile
    for (int idx = tid; idx < 32 * 32; idx += 256) {
        const int orl = idx >> 5, oc = idx & 31;
        const int org = o_r0 + orl;
        const int t0  = (org + 1) & 1;
        const int lrb = ((org + t0 - 5) >> 1) - IR0;
        float acc = 0.0f;
#pragma unroll
        for (int tt = 0; tt < 6; ++tt)
            acc += sh_a[11 - t0 - 2 * tt] * ht[lrb + tt][oc];
        const int ocg = o_c0 + oc;
        if (org < H1 && ocg < W1)
            y1[((size_t)img * H1 + org) * W1 + ocg] = acc;
    }
}

// grid_sample reflection, align_corners=False, size 511
__device__ __forceinline__ float reflect511(float t) {
    t = fabsf(t + 0.5f);
    t = fmodf(t, 1022.0f);
    t = (t >= 511.0f) ? (1022.0f - t) : t;
    t -= 0.5f;
    return fminf(fmaxf(t, 0.0f), 510.0f);
}

// ---------------------------------------------------------------------------
// Stage 2+3 fused: per wave, one 16x16 output tile.
//  - 42x42 warped patch (bilinear, reflect; zero outside 512x512 canvas) -> LDS
//  - horizontal 12-tap decimating conv -> Z[42][16] in LDS
//  - vertical conv as D = Wband(16x42) x Z(42x16) via 11x v_wmma_f32_16x16x4_f32
// ---------------------------------------------------------------------------
__global__ __launch_bounds__(128) void warp_down_kernel(const float* __restrict__ y1,
                                                        const float* __restrict__ Ginv,
                                                        const float* __restrict__ hi2d,
                                                        float* __restrict__ out) {
    __shared__ float sh_a[12];
    __shared__ float shWv[80];        // zero-padded banded-weight row: shWv[32+t]=a[11-t]
    __shared__ float wb[4][42][44];   // warped patch, padded stride
    __shared__ float zb[4][44][16];   // Z + 2 zero pad rows (K padded to 44)

    const int tid  = threadIdx.x;
    const int lane = tid & 31;
    const int wv   = tid >> 5;
    if (tid < 12) sh_a[tid] = hi2d[tid] / sqrtf(hi2d[0]);  // 1D factor (sign-safe)
    if (tid < 80) {
        const int tt = tid - 32;
        float v = 0.0f;
        if (tt >= 0 && tt <= 11) v = hi2d[11 - tt] / sqrtf(hi2d[0]);
        shWv[tid] = v;
    }

    const int T   = blockIdx.x * 4 + wv;    // 49152 tiles total, exact
    const int img = T >> 8;
    const int t   = T & 255;
    const int M0  = (t >> 4) << 4;
    const int N0  = (t & 15) << 4;
    const int b   = img / 3;

    // A = inv(G_inv[b]) via adjugate (bottom row generally (0,0,1) but do full 3x3)
    const float* M = Ginv + b * 9;
    const float m00 = M[0], m01 = M[1], m02 = M[2];
    const float m10 = M[3], m11 = M[4], m12 = M[5];
    const float m20 = M[6], m21 = M[7], m22 = M[8];
    const float c00 = m11 * m22 - m12 * m21;
    const float c01 = m12 * m20 - m10 * m22;
    const float c02 = m10 * m21 - m11 * m20;
    const float inv_det = 1.0f / (m00 * c00 + m01 * c01 + m02 * c02);
    const float a00 = c00 * inv_det, a01 = (m02 * m21 - m01 * m22) * inv_det,
                a02 = (m01 * m12 - m02 * m11) * inv_det;
    const float a10 = c01 * inv_det, a11 = (m00 * m22 - m02 * m20) * inv_det,
                a12 = (m02 * m10 - m00 * m12) * inv_det;
    const float a20 = c02 * inv_det, a21 = (m01 * m20 - m00 * m21) * inv_det,
                a22 = (m00 * m11 - m01 * m10) * inv_det;

    const float* src = y1 + (size_t)img * (H1 * W1);
    __syncthreads();   // sh_a / shWv ready

    // ---- banded A-operand, constant per lane: hoisted out of the WMMA chain.
    // Wv[m][r] = a[11-(r-2m)] for 0 <= r-2m <= 11; A lane layout:
    // lanes 0-15 hold K = k0,k0+1 ; lanes 16-31 hold K = k0+2,k0+3.
    const int m    = lane & 15;
    const int koff = (lane >> 4) * 2;
    const int wbase = koff - 2 * m + 32;   // in [2,34]; max read 75 < 80
    v2f Av[11];
#pragma unroll
    for (int j = 0; j < 11; ++j) {
        Av[j].x = shWv[wbase + 4 * j];
        Av[j].y = shWv[wbase + 4 * j + 1];
    }

    // ---- warped 42x42 patch ----
    for (int idx = lane; idx < 42 * 42; idx += 32) {
        const int r  = idx / 42;
        const int c  = idx - r * 42;
        const int R  = 2 * M0 - 5 + r;   // row on 512x512 warp canvas
        const int Cc = 2 * N0 - 5 + c;   // col
        float val = 0.0f;
        if (R >= 0 && R < 512 && Cc >= 0 && Cc < 512) {
            const float xd = (float)Cc, yd = (float)R;
            const float rw = 1.0f / (a20 * xd + a21 * yd + a22);
            float xq = (a00 * xd + a01 * yd + a02) * rw;
            float yq = (a10 * xd + a11 * yd + a12) * rw;
            xq = reflect511(xq);
            yq = reflect511(yq);
            const float x0 = floorf(xq), y0 = floorf(yq);
            const float wx = xq - x0, wy = yq - y0;
            int x0i = (int)x0; x0i = x0i < 0 ? 0 : (x0i > 510 ? 510 : x0i);
            int y0i = (int)y0; y0i = y0i < 0 ? 0 : (y0i > 510 ? 510 : y0i);
            const int x1i = x0i + 1 > 510 ? 510 : x0i + 1;
            const int y1i = y0i + 1 > 510 ? 510 : y0i + 1;
            const float v00 = src[y0i * W1 + x0i];
            const float v01 = src[y0i * W1 + x1i];
            const float v10 = src[y1i * W1 + x0i];
            const float v11 = src[y1i * W1 + x1i];
            val = v00 * (1.0f - wx) * (1.0f - wy) + v01 * wx * (1.0f - wy)
                + v10 * (1.0f - wx) * wy + v11 * wx * wy;
        }
        wb[wv][r][c] = val;
    }
    __syncthreads();

    // ---- horizontal decimating conv: Z[r][n] = sum_s a[11-s]*wb[r][2n+s] ----
    for (int idx = lane; idx < 42 * 16; idx += 32) {   // 21 per lane, exact
        const int r = idx >> 4, n = idx & 15;
        float acc = 0.0f;
#pragma unroll
        for (int s = 0; s < 12; ++s)
            acc += sh_a[11 - s] * wb[wv][r][2 * n + s];
        zb[wv][r][n] = acc;
    }
    // zero K-pad rows 42,43 (avoid 0*NaN from uninit LDS)
    zb[wv][42 + (lane >> 4)][lane & 15] = 0.0f;
    __syncthreads();

    // ---- batch all B-operand loads, then chain 11 WMMAs on the accumulator ----
    v2f Bv[11];
#pragma unroll
    for (int j = 0; j < 11; ++j) {
        const int k0 = 4 * j + koff;
        Bv[j].x = zb[wv][k0][m];
        Bv[j].y = zb[wv][k0 + 1][m];
    }

    v8f acc = {};   // C = 0
#if __has_builtin(__builtin_amdgcn_wmma_f32_16x16x4_f32)
#pragma unroll
    for (int j = 0; j < 11; ++j)
        acc = __builtin_amdgcn_wmma_f32_16x16x4_f32(false, Av[j], false, Bv[j],
                                                    (short)0, acc, false, false);
#else
    // VALU fallback producing the same C/D lane layout
#pragma unroll
    for (int i = 0; i < 8; ++i) {
        const int mr = (lane >> 4) * 8 + i;
        float s = 0.0f;
        for (int r = 2 * mr; r < 2 * mr + 12; ++r)
            s += sh_a[11 - (r - 2 * mr)] * zb[wv][r][m];
        acc[i] = s;
    }
#endif

    // ---- store per 16x16 f32 C/D layout: VGPR i -> row i + 8*(lane>=16) ----
    const int rbase = (lane >> 4) * 8;
    float* op = out + ((size_t)img * 256 + M0) * 256 + N0;
#pragma unroll
    for (int i = 0; i < 8; ++i)
        op[(size_t)(rbase + i) * 256 + m] = acc[i];
}

extern "C" void kernel_launch(void* const* d_in, const int* in_sizes, int n_in,
                              void* d_out, int out_size, void* d_ws, size_t ws_size,
                              hipStream_t stream) {
    const float* x  = (const float*)d_in[0];   // (64,3,256,256)
    const float* G  = (const float*)d_in[1];   // (64,3,3)
    const float* lo = (const float*)d_in[2];   // (12,12)
    const float* hi = (const float*)d_in[3];   // (12,12)
    float* out = (float*)d_out;
    float* y1  = (float*)d_ws;                 // 192*511*511 f32 = 200.4 MB

    // Stage 1: 192 images x 16x16 tiles of 32x32
    up_lo_kernel<<<IMGS * 256, 256, 0, stream>>>(x, lo, y1);
    // Stage 2+3 fused: 49152 16x16 output tiles, 4 waves/block
    warp_down_kernel<<<IMGS * 256 / 4, 128, 0, stream>>>(y1, G, hi, out);
}